// Model_66322884985434
// MI455X (gfx1250) — compile-verified
//
#include <hip/hip_runtime.h>
#include <math.h>

// ---------------------------------------------------------------------------
// Problem constants (from reference): SEL=20, NN2=10 chunks of L=21,
// NN3=22 chunks of L=20, hidden H=4.
// Output layout (223 floats):
//   [0..32]    sample_probability_array (p0, 10 probs, 22 probs)
//   [33..120]  dW1_2 [4,22]   [121..124] db1_2  [125..128] dW2_2  [129] db2_2
//   [130..213] dW1_3 [4,21]   [214..217] db1_3  [218..221] dW2_3  [222] db2_3
// Workspace layout (floats):
//   ws[0..32]          probs chain
//   ws[64..64+3072)    tables: [set(2)][group(3)][idx(128)][h(4)]
//   ws[3136.. +NBLK*96) per-block partial sums for the current chunk
// ---------------------------------------------------------------------------

#define NBLK 256
#define TPB  256

typedef __attribute__((ext_vector_type(2))) float v2f;
typedef __attribute__((ext_vector_type(8))) float v8f;

// ---------------------------------------------------------------------------
// Table build: T[set][g][idx][h] = sum_j bit_j(idx) * W1[h][1+7g+j]  (+ b1[h]
// for g==0).  Done as 8 M-tiles of 16 rows; K accumulated one bit at a time
// with V_WMMA_F32_16X16X4_F32 rank-1 updates (only K=0 slot populated, which
// sits in VGPR0/lanes0-15 for both A and B — layout-ambiguity-free).
// ---------------------------------------------------------------------------
__global__ void build_tables_kernel(const float* __restrict__ W1_2,
                                    const float* __restrict__ b1_2,
                                    const float* __restrict__ W1_3,
                                    const float* __restrict__ b1_3,
                                    float* __restrict__ tables) {
  const int blk  = blockIdx.x;        // 0..5 = set*3 + g
  const int set  = blk / 3;
  const int g    = blk % 3;
  const int lane = threadIdx.x;       // 32 threads = 1 wave
  const float* W1 = set ? W1_3 : W1_2;
  const float* b1 = set ? b1_3 : b1_2;
  const int L      = set ? 20 : 21;
  const int stride = L + 1;

  // Per-lane weight for B column n=lane (only lanes 0..3 carry hidden units).
  float wj[7];
#pragma unroll
  for (int j = 0; j < 7; ++j) {
    int col = 1 + 7 * g + j;
    float w = 0.0f;
    if (lane < 4 && col <= L) w = W1[lane * stride + col];
    wj[j] = w;
  }

  float* T = tables + (size_t)blk * 128 * 4;

#if __has_builtin(__builtin_amdgcn_wmma_f32_16x16x4_f32)
  for (int t = 0; t < 8; ++t) {
    v8f c = {0.f, 0.f, 0.f, 0.f, 0.f, 0.f, 0.f, 0.f};
    const int mrow = t * 16 + (lane & 15);
#pragma unroll
    for (int j = 0; j < 7; ++j) {
      v2f a;  // A 16x4 f32: VGPR0/lanes0-15 = K=0 column; everything else 0
      a.x = (lane < 16) ? (float)((mrow >> j) & 1) : 0.0f;
      a.y = 0.0f;
      v2f b;  // B 4x16 f32: VGPR0/lanes0-15 = K=0 row; everything else 0
      b.x = (lane < 16) ? wj[j] : 0.0f;
      b.y = 0.0f;
      c = __builtin_amdgcn_wmma_f32_16x16x4_f32(false, a, false, b,
                                                (short)0, c, false, false);
    }
    // D 16x16 f32 layout: VGPR i, lanes 0-15 -> row i, col lane;
    //                     lanes 16-31 -> row i+8, col lane-16.
    const int n = lane & 15;
    if (n < 4) {
      const float badd = (g == 0) ? b1[n] : 0.0f;
#pragma unroll
      for (int i = 0; i < 8; ++i) {
        int m = i + ((lane >> 4) << 3);
        T[(t * 16 + m) * 4 + n] = c[i] + badd;
      }
    }
  }
#else
  // Exact VALU fallback (same results).
  for (int idx = lane; idx < 128; idx += 32) {
#pragma unroll
    for (int h = 0; h < 4; ++h) {
      float s = (g == 0) ? b1[h] : 0.0f;
#pragma unroll
      for (int j = 0; j < 7; ++j) {
        int col = 1 + 7 * g + j;
        float w = (col <= L) ? W1[h * stride + col] : 0.0f;
        s += ((idx >> j) & 1) ? w : 0.0f;
      }
      T[idx * 4 + h] = s;
    }
  }
#endif
}

// ---------------------------------------------------------------------------
// Init: zero the 223 output floats, seed prob chain with p0.
// ---------------------------------------------------------------------------
__global__ void init_kernel(const float* __restrict__ p0,
                            float* __restrict__ out,
                            float* __restrict__ probs) {
  int tid = threadIdx.x;
  if (tid < 223) out[tid] = 0.0f;
  if (tid == 0) {
    float p = p0[0];
    out[0]   = p;
    probs[0] = p;
  }
}

// ---------------------------------------------------------------------------
// Main sweep for one chunk: iterate all 2^L combos, accumulate the 94 masked
// moments.  Deterministic: per-wave shuffle tree -> per-block LDS tree ->
// per-block partials slot (no float atomics anywhere).
// Accumulator map: [0]=output_sum  [1]=gs_b2  [2..5]=gs_W2  [6..9]=gs_b1
//                  [10+h*21+d]=gs_W1[h][1+d]
// ---------------------------------------------------------------------------
__global__ void __launch_bounds__(TPB) sweep_kernel(
    const float* __restrict__ sel, int chunkBase, int L,
    const float* __restrict__ W1, const float* __restrict__ W2,
    const float* __restrict__ b2p, const float* __restrict__ tables,
    const float* __restrict__ probs, int probIdx,
    float* __restrict__ partials) {
  __shared__ float4 Tlds[384];      // 3 groups x 128 entries of [4] = 6 KB
  __shared__ float  red[8][96];

  const float4* Tg = (const float4*)tables;
  for (int i = threadIdx.x; i < 384; i += TPB) Tlds[i] = Tg[i];

  // choices = round(sum(chunk)) -- chunk entries are exact 0.0/1.0
  float csum = 0.0f;
  for (int d = 0; d < L; ++d) csum += sel[chunkBase + d];
  const int choices = (int)rintf(csum);

  const float prob   = probs[probIdx];
  const int   stride = L + 1;
  float w10[4], w2[4];
#pragma unroll
  for (int h = 0; h < 4; ++h) { w10[h] = W1[h * stride]; w2[h] = W2[h]; }
  const float b2 = b2p[0];

  __syncthreads();

  float acc[94];
#pragma unroll
  for (int i = 0; i < 94; ++i) acc[i] = 0.0f;

  const int N       = 1 << L;
  const int gstride = gridDim.x * TPB;
  for (int n = blockIdx.x * TPB + threadIdx.x; n < N; n += gstride) {
    float4 a0 = Tlds[n & 127];
    float4 a1 = Tlds[128 + ((n >> 7) & 127)];
    float4 a2 = Tlds[256 + ((n >> 14) & 127)];
    float h0 = tanhf(fmaf(prob, w10[0], a0.x + a1.x + a2.x));
    float h1 = tanhf(fmaf(prob, w10[1], a0.y + a1.y + a2.y));
    float h2 = tanhf(fmaf(prob, w10[2], a0.z + a1.z + a2.z));
    float h3 = tanhf(fmaf(prob, w10[3], a0.w + a1.w + a2.w));
    float out = fmaf(w2[3], h3, fmaf(w2[2], h2,
                fmaf(w2[1], h1, fmaf(w2[0], h0, b2))));
    const bool  m  = (__popc((unsigned)n) == choices);
    const float e  = expf(out);
    const float wv = m ? out : 0.0f;
    acc[0] += m ? e : 0.0f;
    acc[1] += wv;
    acc[2] = fmaf(wv, h0, acc[2]);
    acc[3] = fmaf(wv, h1, acc[3]);
    acc[4] = fmaf(wv, h2, acc[4]);
    acc[5] = fmaf(wv, h3, acc[5]);
    const float t0 = wv * (1.0f - h0 * h0) * w2[0];
    const float t1 = wv * (1.0f - h1 * h1) * w2[1];
    const float t2 = wv * (1.0f - h2 * h2) * w2[2];
    const float t3 = wv * (1.0f - h3 * h3) * w2[3];
    acc[6] += t0; acc[7] += t1; acc[8] += t2; acc[9] += t3;
#pragma unroll
    for (int d = 0; d < 21; ++d) {   // for L=20, bit 20 of n is always 0
      const float bf = (float)((n >> d) & 1);
      acc[10 + d] = fmaf(bf, t0, acc[10 + d]);
      acc[31 + d] = fmaf(bf, t1, acc[31 + d]);
      acc[52 + d] = fmaf(bf, t2, acc[52 + d]);
      acc[73 + d] = fmaf(bf, t3, acc[73 + d]);
    }
  }

  const int lane = threadIdx.x & 31;
  const int wave = threadIdx.x >> 5;
#pragma unroll
  for (int i = 0; i < 94; ++i) {
    float v = acc[i];
    v += __shfl_xor(v, 16);
    v += __shfl_xor(v, 8);
    v += __shfl_xor(v, 4);
    v += __shfl_xor(v, 2);
    v += __shfl_xor(v, 1);
    if (lane == 0) red[wave][i] = v;
  }
  __syncthreads();
  for (int i = threadIdx.x; i < 94; i += TPB) {
    float s = 0.0f;
#pragma unroll
    for (int w = 0; w < 8; ++w) s += red[w][i];
    partials[(size_t)blockIdx.x * 96 + i] = s;
  }
}

// ---------------------------------------------------------------------------
// Finalize one chunk: fixed-order cross-block reduction, key-combination
// forward pass, p_out, and gradient accumulation into d_out.
// ---------------------------------------------------------------------------
__global__ void finalize_kernel(const float* __restrict__ sel, int chunkBase,
                                int L, const float* __restrict__ W1,
                                const float* __restrict__ b1,
                                const float* __restrict__ W2,
                                const float* __restrict__ b2p,
                                const float* __restrict__ partials, int nblocks,
                                float* __restrict__ probs, int probIdx,
                                float* __restrict__ out, int gradBase) {
  __shared__ float sums[96];
  __shared__ float chunk[21];
  const int tid = threadIdx.x;  // 128 threads

  for (int j = tid; j < 94; j += 128) {
    float s = 0.0f;
    for (int b = 0; b < nblocks; ++b) s += partials[(size_t)b * 96 + j];
    sums[j] = s;
  }
  for (int d = tid; d < L; d += 128) chunk[d] = sel[chunkBase + d];
  __syncthreads();

  const float prob   = probs[probIdx];
  const int   stride = L + 1;
  float hk[4], sk[4];
  float outk = b2p[0];
#pragma unroll
  for (int h = 0; h < 4; ++h) {
    float pre = fmaf(prob, W1[h * stride], b1[h]);
    for (int d = 0; d < L; ++d) pre = fmaf(chunk[d], W1[h * stride + 1 + d], pre);
    hk[h] = tanhf(pre);
    sk[h] = (1.0f - hk[h] * hk[h]) * W2[h];
    outk  = fmaf(W2[h], hk[h], outk);
  }
  const float os  = sums[0];
  const float inv = 1.0f / os;
  const float p   = expf(outk) * inv;
  if (tid == 0) {
    probs[probIdx + 1] = p;
    out[probIdx + 1]   = p;
  }

  const int nW1   = stride * 4;
  const int total = nW1 + 9;
  for (int idx = tid; idx < total; idx += 128) {
    float add;
    int o;
    if (idx < nW1) {
      int h = idx / stride, c = idx % stride;
      float xk   = (c == 0) ? prob : chunk[c - 1];
      float gsum = (c == 0) ? prob * sums[6 + h] : sums[10 + h * 21 + (c - 1)];
      add = sk[h] * xk - gsum * inv;
      o   = gradBase + idx;
    } else if (idx < nW1 + 4) {
      int h = idx - nW1;
      add = sk[h] - sums[6 + h] * inv;
      o   = gradBase + nW1 + h;
    } else if (idx < nW1 + 8) {
      int h = idx - nW1 - 4;
      add = hk[h] - sums[2 + h] * inv;
      o   = gradBase + nW1 + 4 + h;
    } else {
      add = 1.0f - sums[1] * inv;
      o   = gradBase + nW1 + 8;
    }
    out[o] += add;
  }
}

// ---------------------------------------------------------------------------
extern "C" void kernel_launch(void* const* d_in, const int* in_sizes, int n_in,
                              void* d_out, int out_size, void* d_ws,
                              size_t ws_size, hipStream_t stream) {
  (void)in_sizes; (void)n_in; (void)out_size; (void)ws_size;
  const float* p0   = (const float*)d_in[0];
  const float* sel  = (const float*)d_in[1];
  const float* W1_2 = (const float*)d_in[2];
  const float* b1_2 = (const float*)d_in[3];
  const float* W2_2 = (const float*)d_in[4];
  const float* b2_2 = (const float*)d_in[5];
  const float* W1_3 = (const float*)d_in[6];
  const float* b1_3 = (const float*)d_in[7];
  const float* W2_3 = (const float*)d_in[8];
  const float* b2_3 = (const float*)d_in[9];

  float* out      = (float*)d_out;
  float* ws       = (float*)d_ws;
  float* probs    = ws;              // 33 floats (padded to 64)
  float* tables   = ws + 64;         // 2*3*128*4 = 3072 floats
  float* partials = ws + 64 + 3072;  // NBLK*96 floats

  hipLaunchKernelGGL(init_kernel, dim3(1), dim3(256), 0, stream, p0, out, probs);
  hipLaunchKernelGGL(build_tables_kernel, dim3(6), dim3(32), 0, stream,
                     W1_2, b1_2, W1_3, b1_3, tables);

  for (int i = 0; i < 32; ++i) {
    const int set = (i < 10) ? 0 : 1;
    const int L = set ? 20 : 21;
    const int base = set ? (210 + (i - 10) * 20) : (i * 21);
    const float* W1 = set ? W1_3 : W1_2;
    const float* b1 = set ? b1_3 : b1_2;
    const float* W2 = set ? W2_3 : W2_2;
    const float* b2 = set ? b2_3 : b2_2;
    const float* T  = tables + set * 1536;
    const int gradBase = set ? 130 : 33;

    hipLaunchKernelGGL(sweep_kernel, dim3(NBLK), dim3(TPB), 0, stream,
                       sel, base, L, W1, W2, b2, T, probs, i, partials);
    hipLaunchKernelGGL(finalize_kernel, dim3(1), dim3(128), 0, stream,
                       sel, base, L, W1, b1, W2, b2, partials, NBLK,
                       probs, i, out, gradBase);
  }
}